// MaxKGINConv_11768210391440
// MI455X (gfx1250) — compile-verified
//
#include <hip/hip_runtime.h>
#include <hip/hip_bf16.h>

typedef __attribute__((ext_vector_type(2))) float v2f;
typedef __attribute__((ext_vector_type(8))) float v8f;

#define DIM 128
#define KSEL 32

// ---------------------------------------------------------------------------
// Kernel 1: per-row MaxK threshold + sparsify + zero-init the neigh buffer.
// One wave32 per row; each lane owns 4 of the 128 features.
// Exact k-th largest via 32 rounds of (wave max, remove one instance).
// ---------------------------------------------------------------------------
__global__ void __launch_bounds__(256) maxk_sparsify_kernel(
    const float* __restrict__ feat,
    float* __restrict__ sparse,
    float* __restrict__ neigh,
    int N)
{
    const int wave = threadIdx.x >> 5;
    const int lane = threadIdx.x & 31;
    const int row  = blockIdx.x * 8 + wave;
    if (row >= N) return;

    const float4 v = reinterpret_cast<const float4*>(feat + (size_t)row * DIM)[lane];
    float x[4] = {v.x, v.y, v.z, v.w};
    bool  alive[4] = {true, true, true, true};
    float thresh = 0.0f;

    #pragma unroll 1
    for (int it = 0; it < KSEL; ++it) {
        float lm = -__builtin_inff();
        int   li = -1;
        #pragma unroll
        for (int i = 0; i < 4; ++i)
            if (alive[i] && x[i] > lm) { lm = x[i]; li = i; }
        // wave32 max reduction
        float wm = lm;
        #pragma unroll
        for (int off = 16; off >= 1; off >>= 1)
            wm = fmaxf(wm, __shfl_xor(wm, off, 32));
        // remove exactly one instance of the max (lowest lane wins)
        unsigned long long ball = __ballot((li >= 0) && (lm == wm));
        int winner = (int)__ffsll(ball) - 1;
        if (lane == winner) alive[li] = false;
        thresh = wm;
    }

    float4 s;
    s.x = (v.x >= thresh) ? v.x : 0.0f;
    s.y = (v.y >= thresh) ? v.y : 0.0f;
    s.z = (v.z >= thresh) ? v.z : 0.0f;
    s.w = (v.w >= thresh) ? v.w : 0.0f;
    reinterpret_cast<float4*>(sparse + (size_t)row * DIM)[lane] = s;

    float4 z; z.x = 0.0f; z.y = 0.0f; z.z = 0.0f; z.w = 0.0f;
    reinterpret_cast<float4*>(neigh + (size_t)row * DIM)[lane] = z;
}

// ---------------------------------------------------------------------------
// Kernel 2: edge scatter-add. One wave32 per edge, 4 floats per lane.
// Skip atomics on zeros (only 32/128 entries survive MaxK) -> 4x fewer
// L2 atomic ops. Both sparse and neigh are L2-resident (25.6 MB each).
// ---------------------------------------------------------------------------
__global__ void __launch_bounds__(256) scatter_edges_kernel(
    const float* __restrict__ sparse,
    const int* __restrict__ esrc,
    const int* __restrict__ edst,
    float* __restrict__ neigh,
    int E)
{
    const int wave = threadIdx.x >> 5;
    const int lane = threadIdx.x & 31;
    const int e    = blockIdx.x * 8 + wave;
    if (e >= E) return;

    const int s = esrc[e];
    const int d = edst[e];

    const float4 m = reinterpret_cast<const float4*>(sparse + (size_t)s * DIM)[lane];
    float* out = neigh + (size_t)d * DIM + lane * 4;

    if (m.x != 0.0f) atomicAdd(out + 0, m.x);
    if (m.y != 0.0f) atomicAdd(out + 1, m.y);
    if (m.z != 0.0f) atomicAdd(out + 2, m.z);
    if (m.w != 0.0f) atomicAdd(out + 3, m.w);
}

// ---------------------------------------------------------------------------
// Kernel 3: Out = act(A @ W^T + bias), A is [N,128], W is [128,128] row-major.
// FUSE_ADD: A = (1+eps)*A0 + A1 (built on the fly while staging into LDS).
// Full fp32 WMMA: V_WMMA_F32_16X16X4_F32, wave32, one 16x16 tile per wave,
// 8 waves/block -> block covers 16 rows x 128 cols. 32 WMMAs per tile (K=128).
// ---------------------------------------------------------------------------
template <bool FUSE_ADD, bool RELU>
__global__ void __launch_bounds__(256) gemm_wmma_f32_kernel(
    const float* __restrict__ A0,
    const float* __restrict__ A1,    // neigh (FUSE_ADD) or nullptr
    const float* __restrict__ eps,   // scalar (FUSE_ADD) or nullptr
    const float* __restrict__ W,     // [128,128] row-major
    const float* __restrict__ bias,  // [128]
    float* __restrict__ Out,
    int N)
{
    __shared__ float As[16][132];    // pad 128->132: conflict-free column reads

    const int row0 = blockIdx.x * 16;
    const int wave = threadIdx.x >> 5;
    const int lane = threadIdx.x & 31;

    const float alpha = FUSE_ADD ? (1.0f + eps[0]) : 1.0f;

    // Stage 16x128 A-tile into LDS (2048 floats, 256 threads, float2 each x4).
    {
        const int t = threadIdx.x;
        #pragma unroll
        for (int i = 0; i < 4; ++i) {
            const int idx = t + i * 256;       // float2 index in [0,1024)
            const int r   = idx >> 6;          // 64 float2 per row
            const int c   = (idx & 63) * 2;
            const size_t g = (size_t)(row0 + r) * DIM + c;
            float2 a = *reinterpret_cast<const float2*>(A0 + g);
            if (FUSE_ADD) {
                const float2 n2 = *reinterpret_cast<const float2*>(A1 + g);
                a.x = alpha * a.x + n2.x;
                a.y = alpha * a.y + n2.y;
            }
            As[r][c]     = a.x;
            As[r][c + 1] = a.y;
        }
    }
    __syncthreads();

    // WMMA f32 16x16x4 operand mapping (wave32):
    //   A 16x4 : lanes 0-15 -> M=lane, K pair {0,1}; lanes 16-31 -> K pair {2,3}
    //   B 4x16 : lanes 0-15 -> N=lane, K pair {0,1}; lanes 16-31 -> K pair {2,3}
    const int n0 = wave * 16;
    const int m  = lane & 15;              // doubles as local N index for B
    const int kb = (lane >> 4) << 1;       // 0 or 2
    const float* Wrow = W + (size_t)(n0 + m) * DIM;

    v8f acc = {};
    #pragma unroll
    for (int k = 0; k < DIM; k += 4) {
        v2f a = *reinterpret_cast<const v2f*>(&As[m][k + kb]);
        v2f b = *reinterpret_cast<const v2f*>(Wrow + k + kb);
        acc = __builtin_amdgcn_wmma_f32_16x16x4_f32(
                  /*neg_a=*/false, a, /*neg_b=*/false, b,
                  /*c_mod=*/(short)0, acc, /*reuse_a=*/false, /*reuse_b=*/false);
    }

    // C/D layout: VGPR r -> M=r (lanes 0-15) or M=r+8 (lanes 16-31), N=lane&15.
    const int   nn    = n0 + m;
    const float bn    = bias[nn];
    const int   mbase = (lane >> 4) * 8;
    #pragma unroll
    for (int r = 0; r < 8; ++r) {
        float vv = acc[r] + bn;
        if (RELU) vv = fmaxf(vv, 0.0f);
        Out[(size_t)(row0 + mbase + r) * DIM + nn] = vv;
    }
}

// ---------------------------------------------------------------------------
// Launch. Inputs: feat, W1, b1, W2, b2, eps, edge_src, edge_dst.
// Workspace: sparse[N*128] | neigh[N*128] | h[N*128]  (~77 MB fp32).
// ---------------------------------------------------------------------------
extern "C" void kernel_launch(void* const* d_in, const int* in_sizes, int n_in,
                              void* d_out, int out_size, void* d_ws, size_t ws_size,
                              hipStream_t stream) {
    const float* feat = (const float*)d_in[0];
    const float* W1   = (const float*)d_in[1];
    const float* b1   = (const float*)d_in[2];
    const float* W2   = (const float*)d_in[3];
    const float* b2   = (const float*)d_in[4];
    const float* eps  = (const float*)d_in[5];
    const int*   esrc = (const int*)d_in[6];
    const int*   edst = (const int*)d_in[7];

    const int N = in_sizes[0] / DIM;
    const int E = in_sizes[6];

    float* out    = (float*)d_out;
    float* sparse = (float*)d_ws;
    float* neigh  = sparse + (size_t)N * DIM;
    float* h      = neigh  + (size_t)N * DIM;

    // 1) threshold + sparsify + zero neigh
    maxk_sparsify_kernel<<<(N + 7) / 8, 256, 0, stream>>>(feat, sparse, neigh, N);

    // 2) edge scatter-add
    scatter_edges_kernel<<<(E + 7) / 8, 256, 0, stream>>>(sparse, esrc, edst, neigh, E);

    // 3) h = relu(((1+eps)*feat + neigh) @ W1^T + b1)   [N divisible by 16]
    gemm_wmma_f32_kernel<true, true><<<N / 16, 256, 0, stream>>>(
        feat, neigh, eps, W1, b1, h, N);

    // 4) out = h @ W2^T + b2
    gemm_wmma_f32_kernel<false, false><<<N / 16, 256, 0, stream>>>(
        h, nullptr, nullptr, W2, b2, out, N);
}